// Abstract_GCN_54176717472164
// MI455X (gfx1250) — compile-verified
//
#include <hip/hip_runtime.h>
#include <stdint.h>

typedef __attribute__((ext_vector_type(2))) float v2f;
typedef __attribute__((ext_vector_type(8))) float v8f;

#define D 128   // D_IN == D_OUT == 128

// ---------------------------------------------------------------------------
// Kernel 0: zero the workspace (agg + deg). Must run every call: the harness
// poisons d_ws once and never re-poisons, and graph replays reuse it.
// ---------------------------------------------------------------------------
__global__ void gcn_zero_ws(float* __restrict__ ws, long long n) {
  long long i = (long long)blockIdx.x * blockDim.x + threadIdx.x;
  long long stride = (long long)gridDim.x * blockDim.x;
  for (; i < n; i += stride) ws[i] = 0.0f;
}

// ---------------------------------------------------------------------------
// Kernel 1: degree = float bincount(row). L2-resident float atomics.
// ---------------------------------------------------------------------------
__global__ void gcn_degree(const long long* __restrict__ row,
                           float* __restrict__ deg, int E) {
  int e = blockIdx.x * blockDim.x + threadIdx.x;
  if (e < E) atomicAdd(&deg[row[e]], 1.0f);
}

// ---------------------------------------------------------------------------
// Kernel 2: agg[row[e], :] += x[col[e], :].
// One wave32 per edge: lanes cover 32 consecutive features -> 128B coalesced
// reads of x[col] and coalesced global_atomic_add_f32 into agg[row].
// Both x (25.6MB) and agg (25.6MB) are L2-resident (192MB L2).
// ---------------------------------------------------------------------------
__global__ void gcn_scatter(const float* __restrict__ x,
                            const long long* __restrict__ row,
                            const long long* __restrict__ col,
                            float* __restrict__ agg, int E) {
  int e = blockIdx.x * (blockDim.x >> 5) + (threadIdx.x >> 5);  // wave-uniform
  int lane = threadIdx.x & 31;
  if (e >= E) return;
  long long r = row[e];
  long long c = col[e];
  const float* xs = x + c * D;
  float* as = agg + r * D;
#pragma unroll
  for (int i = 0; i < 4; ++i) {
    int f = lane + i * 32;
    atomicAdd(&as[f], xs[f]);
  }
}

// ---------------------------------------------------------------------------
// Kernel 3: out = (agg @ W^T) * (1/deg) + b using V_WMMA_F32_16X16X4_F32.
// (Row-scaling commutes with the GEMM, so the mean-normalize is fused into
//  the epilogue as an FMA; this lets the A tile be DMA'd raw into LDS and
//  avoids per-element IEEE divide chains — only 16 reciprocals per block.)
//
// Block = 256 threads = 8 waves. Block owns one 16-node tile (m0), wave w owns
// output-feature tile j0 = 16*w. The 16x128 agg tile (8KB) is staged into LDS
// with GLOBAL_LOAD_ASYNC_TO_LDS_B128 (ASYNCcnt path, no VGPR round-trip):
// 256 threads x 2 x 16B. W fragments come from global (64KB W is L0/L2
// resident across all 3125 blocks).
//
// Fragment layouts (wave32, ISA 7.12.2):
//   A 16x4 f32:  lane m = lane&15, VGPR v holds K = v + 2*(lane>>4)
//   B 4x16 f32:  lane n = lane&15, VGPR v holds K = v + 2*(lane>>4)
//   C/D 16x16:   lane n = lane&15, VGPR v holds row M = v + 8*(lane>>4)
// ---------------------------------------------------------------------------
__global__ void __launch_bounds__(256) gcn_gemm_wmma(
    const float* __restrict__ agg, const float* __restrict__ deg,
    const float* __restrict__ W, const float* __restrict__ bias,
    float* __restrict__ out, int n_nodes) {
  __shared__ float sA[16 * D];   // raw agg tile, 8KB
  __shared__ float sRDeg[16];    // per-row reciprocal degree

  const int tid = threadIdx.x;
  const int m0 = blockIdx.x * 16;

  // --- async DMA of the 16x128 agg tile into LDS (2 x b128 per thread) ---
  {
    uint32_t lds0 = (uint32_t)(uintptr_t)(&sA[0]) + (uint32_t)tid * 16u;
    uint32_t lds1 = lds0 + 4096u;                       // second half of tile
    uint32_t voff = (uint32_t)tid * 16u;                // byte offset, GVS mode
    unsigned long long gbase =
        (unsigned long long)(uintptr_t)(agg + (size_t)m0 * D);
    asm volatile("global_load_async_to_lds_b128 %0, %1, %2"
                 :: "v"(lds0), "v"(voff), "s"(gbase) : "memory");
    asm volatile("global_load_async_to_lds_b128 %0, %1, %2 offset:4096"
                 :: "v"(lds1), "v"(voff), "s"(gbase) : "memory");
  }
  // reciprocal degrees for the 16 rows of this tile (one divide per row)
  if (tid < 16) sRDeg[tid] = 1.0f / deg[m0 + tid];
  asm volatile("s_wait_asynccnt 0x0" ::: "memory");
  __syncthreads();

  const int wave = tid >> 5;     // 0..7 -> output feature tile
  const int lane = tid & 31;
  const int half = lane >> 4;    // 0 or 1
  const int l16  = lane & 15;
  const int j0   = wave * 16;

  v8f c;
#pragma unroll
  for (int v = 0; v < 8; ++v) c[v] = 0.0f;

  const float* wrow = W + (size_t)(j0 + l16) * D;  // B[k][n] = W[j0+n][k]
  const float* arow = sA + l16 * D;

#pragma unroll 4
  for (int k = 0; k < D; k += 4) {
    const int kb = k + 2 * half;
    v2f a, b2;
    a.x  = arow[kb];         // A: K = 0 + 2*half
    a.y  = arow[kb + 1];     // A: K = 1 + 2*half
    b2.x = wrow[kb];         // B: K = 0 + 2*half
    b2.y = wrow[kb + 1];     // B: K = 1 + 2*half
    // (neg_a, A, neg_b, B, c_mod, C, reuse_a, reuse_b)
    c = __builtin_amdgcn_wmma_f32_16x16x4_f32(
        false, a, false, b2, (short)0, c, false, false);
  }

  // Epilogue: out[node, j] = c * rdeg[node] + b[j]  (one FMA per element).
  // N is a multiple of 16 and the grid covers it exactly -> no guards, 8
  // coalesced global_store_b32 (lanes span 16 consecutive columns per half).
  const float bv = bias[j0 + l16];
  float* obase = out + (size_t)m0 * D + (size_t)(j0 + l16);
#pragma unroll
  for (int v = 0; v < 8; ++v) {
    int r = v + 8 * half;
    obase[(size_t)r * D] = fmaf(c[v], sRDeg[r], bv);
  }
}

// ---------------------------------------------------------------------------
extern "C" void kernel_launch(void* const* d_in, const int* in_sizes, int n_in,
                              void* d_out, int out_size, void* d_ws, size_t ws_size,
                              hipStream_t stream) {
  const float* x          = (const float*)d_in[0];      // [N, 128] fp32
  const long long* edges  = (const long long*)d_in[1];  // [2, E] int64
  const float* W          = (const float*)d_in[2];      // [128, 128] fp32
  const float* bias       = (const float*)d_in[3];      // [128] fp32
  float* out              = (float*)d_out;              // [N, 128] fp32

  const int n_nodes = in_sizes[0] / D;
  const int E       = in_sizes[1] / 2;
  const long long* row = edges;        // edge_index[0]
  const long long* col = edges + E;    // edge_index[1]

  // Workspace layout: agg [N*128] fp32, then deg [N] fp32 (contiguous).
  float* agg = (float*)d_ws;
  float* deg = agg + (size_t)n_nodes * D;

  // 1) zero agg+deg
  long long zn = (long long)n_nodes * D + n_nodes;
  gcn_zero_ws<<<2048, 256, 0, stream>>>(agg, zn);

  // 2) degrees
  gcn_degree<<<(E + 255) / 256, 256, 0, stream>>>(row, deg, E);

  // 3) edge scatter-add (one wave32 per edge, 8 edges per 256-thread block)
  int sblocks = (E + 7) / 8;
  gcn_scatter<<<sblocks, 256, 0, stream>>>(x, row, col, agg, E);

  // 4) WMMA GEMM: (agg @ W^T) * (1/deg) + b
  int mtiles = (n_nodes + 15) / 16;
  gcn_gemm_wmma<<<mtiles, 256, 0, stream>>>(agg, deg, W, bias, out, n_nodes);
}